// LossMIDU_16080357556378
// MI455X (gfx1250) — compile-verified
//
#include <hip/hip_runtime.h>

#define EDGE 4096
#define NCELL (EDGE * EDGE)

typedef __attribute__((ext_vector_type(2))) float v2f;
typedef __attribute__((ext_vector_type(8))) float v8f;

// ---------------- lock-free union-find helpers ----------------

__device__ __forceinline__ int find_root(const int* L, int x) {
    // parents only ever decrease (atomicMin links hi->lo), so plain volatile
    // chasing terminates at the component's minimum index.
    const volatile int* vL = (const volatile int*)L;
    int p = vL[x];
    while (p != x) {
        x = p;
        p = vL[x];
    }
    return x;
}

__device__ __forceinline__ void unite(int* L, int a, int b) {
    int ra = find_root(L, a);
    int rb = find_root(L, b);
    while (ra != rb) {
        int lo = ra < rb ? ra : rb;
        int hi = ra < rb ? rb : ra;
        int old = atomicMin(&L[hi], lo);
        if (old == hi) break;          // hi was a root, now linked to lo
        ra = find_root(L, lo);         // hi already had parent 'old': merge lo with old
        rb = find_root(L, old);
    }
}

// ---------------- WMMA-based full-wave32 sum ----------------
// A is 16x4 f32 (per-lane {t,0}; lanes 0-15 hold K=0,1 / lanes 16-31 hold K=2,3).
// B = ones(4x16)  =>  D[m][n] = t[m] + t[m+16].
// C/D layout: lane n (n<16) holds D[0..7][n] in c[0..7]; lane n>=16 holds D[8..15][n-16].
// So sum(c[0..7]) = lanes {0..7,16..23} on the low half, lanes {8..15,24..31} on the
// high half; one xor-16 shuffle-add completes the full 32-lane sum (on every lane).
__device__ __forceinline__ float wave_sum_wmma(float t) {
    v2f a; a.x = t;   a.y = 0.0f;
    v2f b; b.x = 1.0f; b.y = 1.0f;
    v8f c = {};
    c = __builtin_amdgcn_wmma_f32_16x16x4_f32(
            /*neg_a=*/false, a, /*neg_b=*/false, b,
            /*c_mod=*/(short)0, c, /*reuse_a=*/false, /*reuse_b=*/false);
    float s = c[0] + c[1] + c[2] + c[3] + c[4] + c[5] + c[6] + c[7];
    s += __shfl_xor(s, 16);
    return s;
}

// ---------------- kernels ----------------

__global__ void k_init(const float* __restrict__ x, int* __restrict__ L,
                       float* __restrict__ sums, int* __restrict__ counts,
                       float* __restrict__ scal) {
    int i = blockIdx.x * blockDim.x + threadIdx.x;
    if (i == 0) { scal[0] = 0.0f; scal[1] = 0.0f; }
    if (i < NCELL) {
        L[i] = i;            // every cell its own root; only masked cells get merged
        sums[i] = 0.0f;
        counts[i] = 0;
    }
}

__global__ void k_merge(const float* __restrict__ x, int* __restrict__ L) {
    int i = blockIdx.x * blockDim.x + threadIdx.x;
    if (i >= NCELL) return;
    if (!(x[i] > 0.0f)) return;                 // tanh(x)>0  <=>  x>0
    int c = i & (EDGE - 1);
    if (c + 1 < EDGE && x[i + 1] > 0.0f)        unite(L, i, i + 1);     // right
    if (i + EDGE < NCELL && x[i + EDGE] > 0.0f) unite(L, i, i + EDGE);  // down
}

__global__ void k_accum(const float* __restrict__ x, const int* __restrict__ L,
                        float* __restrict__ sums, int* __restrict__ counts) {
    int i = blockIdx.x * blockDim.x + threadIdx.x;
    if (i >= NCELL) return;
    float xv = x[i];
    if (!(xv > 0.0f)) return;
    int root = find_root(L, i);                 // root == min index of component
    atomicAdd(&sums[root], tanhf(xv));
    atomicAdd(&counts[root], 1);
}

__global__ void k_reduce(const float* __restrict__ sums, const int* __restrict__ counts,
                         float* __restrict__ scal) {
    float t_acc = 0.0f;   // sum over components of sums/(N+1-count)
    float n_acc = 0.0f;   // number of components
    int stride = gridDim.x * blockDim.x;
    for (int i = blockIdx.x * blockDim.x + threadIdx.x; i < NCELL; i += stride) {
        int cnt = counts[i];
        if (cnt > 0) {
            t_acc += sums[i] / (float)(NCELL + 1 - cnt);
            n_acc += 1.0f;
        }
    }
    // EXEC is all-ones here (no divergent exits), as required for WMMA.
    float tw = wave_sum_wmma(t_acc);
    float nw = wave_sum_wmma(n_acc);

    __shared__ float sh_t[8];
    __shared__ float sh_n[8];
    int lane = threadIdx.x & 31;
    int wave = threadIdx.x >> 5;
    if (lane == 0) { sh_t[wave] = tw; sh_n[wave] = nw; }
    __syncthreads();
    if (threadIdx.x == 0) {
        float T = 0.0f, C = 0.0f;
        int nwaves = blockDim.x >> 5;
        for (int w = 0; w < nwaves; ++w) { T += sh_t[w]; C += sh_n[w]; }
        atomicAdd(&scal[0], T);
        atomicAdd(&scal[1], C);
    }
}

__global__ void k_final(const float* __restrict__ scal, float* __restrict__ out) {
    if (blockIdx.x == 0 && threadIdx.x == 0) {
        out[0] = (scal[1] > 0.0f) ? (scal[0] / scal[1]) : 0.0f;
    }
}

// ---------------- launch ----------------

extern "C" void kernel_launch(void* const* d_in, const int* in_sizes, int n_in,
                              void* d_out, int out_size, void* d_ws, size_t ws_size,
                              hipStream_t stream) {
    (void)in_sizes; (void)n_in; (void)out_size; (void)ws_size;

    const float* x = (const float*)d_in[0];
    char* ws = (char*)d_ws;
    int*   L      = (int*)(ws);                                  //  64 MB
    float* sums   = (float*)(ws + (size_t)NCELL * 4);            //  64 MB
    int*   counts = (int*)(ws + (size_t)NCELL * 8);              //  64 MB
    float* scal   = (float*)(ws + (size_t)NCELL * 12);           //  8 B
    float* out    = (float*)d_out;

    const int BLK = 256;                       // 8 wave32s per block
    const int GRD = (NCELL + BLK - 1) / BLK;

    k_init  <<<GRD,  BLK, 0, stream>>>(x, L, sums, counts, scal);
    k_merge <<<GRD,  BLK, 0, stream>>>(x, L);
    k_accum <<<GRD,  BLK, 0, stream>>>(x, L, sums, counts);
    k_reduce<<<4096, BLK, 0, stream>>>(sums, counts, scal);
    k_final <<<1,    32,  0, stream>>>(scal, out);
}